// QuantumGenerator_893353197950
// MI455X (gfx1250) — compile-verified
//
#include <hip/hip_runtime.h>
#include <math.h>

#define NQ      16
#define NB      64
#define NSTATE  (1 << NQ)          // 65536 amplitudes / sample
#define QDEPTH  4
#define IMGPIX  784                // 28*28
#define HALFST  (NSTATE / 2)       // 32768 amps, 256 KB LDS-resident chunk

// ---- static device buffers (fully overwritten before any read, every call) ----
__device__ float2 g_stA[NB * NSTATE];   // 32 MB state buffer A
__device__ float2 g_stB[NB * NSTATE];   // 32 MB state buffer B
__device__ float  g_cz[NB * NQ], g_sz[NB * NQ];         // cos/sin(z/2) per sample/qubit
__device__ float  g_cw[QDEPTH * NQ], g_sw[QDEPTH * NQ]; // cos/sin(w/2) per layer/qubit
__device__ float  g_zq[NB * NQ];        // <Z_q> expvals

// WMMA vector types (f32 16x16x4: A/B = 2 VGPRs, C/D = 8 VGPRs per lane)
typedef float v2f __attribute__((ext_vector_type(2)));
typedef float v8f __attribute__((ext_vector_type(8)));

// TDM descriptor vector types
typedef unsigned int u32x4 __attribute__((ext_vector_type(4)));
typedef int          i32x4 __attribute__((ext_vector_type(4)));
typedef int          i32x8 __attribute__((ext_vector_type(8)));

// Toolchain detection: therock-10.0 headers ship the gfx1250 TDM header and the
// 6-arg builtin; ROCm 7.2 has the 5-arg builtin.
#if defined(__has_include)
#if __has_include(<hip/amd_detail/amd_gfx1250_TDM.h>)
#define TDM_SIX_ARG 1
#endif
#endif

#if defined(__has_builtin)
#if __has_builtin(__builtin_amdgcn_tensor_load_to_lds) && \
    __has_builtin(__builtin_amdgcn_tensor_store_from_lds)
#define HAVE_TDM 1
#endif
#endif

// ---------------------------------------------------------------------------
// 0) Angle tables: cos/sin of half-angles for z (64x16) and layer weights (4x16)
// ---------------------------------------------------------------------------
__global__ void k_tables(const float* __restrict__ z, const float* __restrict__ w) {
    int t = threadIdx.x;
    if (t < NB * NQ) {
        float a = 0.5f * z[t];
        g_cz[t] = __cosf(a);
        g_sz[t] = __sinf(a);
    }
    if (t < QDEPTH * NQ) {
        float a = 0.5f * w[t];
        g_cw[t] = __cosf(a);
        g_sw[t] = __sinf(a);
    }
}

// ---------------------------------------------------------------------------
// 1) Initial state in closed form (|0..0> followed by RX(z_q) on every qubit):
//    amp(x) = [prod_q (x_q ? sin : cos)] * (-i)^popcount(x)
//    Index convention: bit q of flat index == qubit q.
//    Layout: x = (j<<10) | tid  -> coalesced float2 stores.
// ---------------------------------------------------------------------------
__global__ void k_init() {
    const int s   = blockIdx.x;
    const int tid = threadIdx.x;           // 0..1023 -> qubits 0..9
    float cq[NQ], sq[NQ];
#pragma unroll
    for (int q = 0; q < NQ; ++q) {
        cq[q] = g_cz[s * NQ + q];
        sq[q] = g_sz[s * NQ + q];
    }
    float PL = 1.0f;
#pragma unroll
    for (int b = 0; b < 10; ++b)
        PL *= ((tid >> b) & 1) ? sq[b] : cq[b];
    const int pcT = __popc(tid);
#pragma unroll
    for (int j = 0; j < 64; ++j) {         // qubits 10..15
        float PH = 1.0f;
#pragma unroll
        for (int b = 0; b < 6; ++b)
            PH *= ((j >> b) & 1) ? sq[10 + b] : cq[10 + b];
        const float P = PL * PH;
        const int k = (pcT + __popc(j)) & 3;     // (-i)^k
        const float re = (k == 0) ? P : ((k == 2) ? -P : 0.0f);
        const float im = (k == 1) ? -P : ((k == 3) ? P : 0.0f);
        g_stA[(size_t)s * NSTATE + (j << 10) + tid] = make_float2(re, im);
    }
}

// ---------------------------------------------------------------------------
// 2) One entangler layer, qubits 0..14: 15 RX butterfly stages fully in LDS.
//    Block = half a sample (32768 amps = 256 KB LDS, 1 block per WGP).
//    LDS fill/drain via the Tensor Data Mover: one TENSOR_LOAD_TO_LDS /
//    TENSOR_STORE_FROM_LDS per block (1-D 256 KB tile, data_size=8B),
//    tracked with TENSORcnt.
//    RX combine (side-independent): new = c*mine + (-i*s)*partner
//      re' = c*a.re + s*b.im ; im' = c*a.im - s*b.re
// ---------------------------------------------------------------------------
__global__ void k_layer(int layer, int sel) {
    extern __shared__ float2 st[];                 // 32768 float2
    float2* gs = sel ? g_stB : g_stA;
    const int s = blockIdx.x >> 1;
    const int h = blockIdx.x & 1;                  // bit 15 of flat index
    const size_t base = (size_t)s * NSTATE + (size_t)h * HALFST;
    const int tid = threadIdx.x;                   // 1024 threads

#if defined(HAVE_TDM)
    // ---- Tensor DMA descriptor (D#): 1-D tile of 32768 x 8B elements ----
    const unsigned long long gaddr = (unsigned long long)(const void*)(gs + base);
    // addrspace(3) flat aperture: low 32 bits of the flat address are the raw
    // LDS byte offset.
    const unsigned ldsaddr = (unsigned)(unsigned long long)(void*)st;

    u32x4 g0;
    g0[0] = 1u;                                   // count=1, user load descriptor
    g0[1] = ldsaddr;                              // lds_addr (bytes)
    g0[2] = (unsigned)(gaddr & 0xFFFFFFFFull);    // global_addr[31:0]
    g0[3] = (unsigned)((gaddr >> 32) & 0x01FFFFFFull) | (2u << 30); // [56:32] | type=2

    i32x8 g1;
    g1[0] = 0x00030000;            // workgroup_mask=0, data_size=3 (8 bytes)
    g1[1] = (int)(HALFST << 16);   // tensor_dim0[15:0] in bits 63:48 (=32768 lo16=0x8000... )
    g1[2] = (int)(1u << 16);       // tensor_dim0[31:16]=0 | tensor_dim1[15:0]=1
    g1[3] = (int)0x80000000u;      // tensor_dim1[31:16]=0 | tile_dim0=32768
    g1[4] = 1;                     // tile_dim1=1, tile_dim2=0
    g1[5] = HALFST;                // tensor_dim0_stride[31:0]
    g1[6] = 0;                     // stride hi | tensor_dim1_stride lo
    g1[7] = 0;

    i32x4 g2 = {0, 0, 0, 0};
    i32x4 g3 = {0, 0, 0, 0};
#if defined(TDM_SIX_ARG)
    i32x8 g4 = {0, 0, 0, 0, 0, 0, 0, 0};
#endif

    if (tid == 0) {                // TDM ignores EXEC; issue once per block
#if defined(TDM_SIX_ARG)
        __builtin_amdgcn_tensor_load_to_lds(g0, g1, g2, g3, g4, 0);
#else
        __builtin_amdgcn_tensor_load_to_lds(g0, g1, g2, g3, 0);
#endif
        __builtin_amdgcn_s_wait_tensorcnt(0);
    }
    __syncthreads();
#else
#pragma unroll
    for (int i = 0; i < 32; ++i)
        st[i * 1024 + tid] = gs[base + i * 1024 + tid];
    __syncthreads();
#endif

    for (int q = 0; q < 15; ++q) {
        const float c  = g_cw[layer * NQ + q];
        const float sn = g_sw[layer * NQ + q];
        const int m = 1 << q;
#pragma unroll
        for (int p = 0; p < 16; ++p) {             // 16384 pairs / 1024 threads
            const int pid = p * 1024 + tid;
            const int x0 = ((pid >> q) << (q + 1)) | (pid & (m - 1));
            const int x1 = x0 | m;
            const float2 a = st[x0];
            const float2 b = st[x1];
            st[x0] = make_float2(c * a.x + sn * b.y, c * a.y - sn * b.x);
            st[x1] = make_float2(c * b.x + sn * a.y, c * b.y - sn * a.x);
        }
        __syncthreads();
    }

#if defined(HAVE_TDM)
    if (tid == 0) {
#if defined(TDM_SIX_ARG)
        __builtin_amdgcn_tensor_store_from_lds(g0, g1, g2, g3, g4, 0);
#else
        __builtin_amdgcn_tensor_store_from_lds(g0, g1, g2, g3, 0);
#endif
        __builtin_amdgcn_s_wait_tensorcnt(0);
    }
#else
#pragma unroll
    for (int i = 0; i < 32; ++i)
        gs[base + i * 1024 + tid] = st[i * 1024 + tid];
#endif
}

// ---------------------------------------------------------------------------
// 3) Qubit-15 RX fused with the CNOT-ring permutation.
//    Ring CNOT(q->q+1 mod 16) composite (forward): y0 = x1^..^x15, y_q = x0^..^x_q.
//    Inverse: x0 = y0^y15 ; x1 = y0^y1^y15 ; x_q = y_{q-1}^y_q (q>=2).
//    out[y] = c*in[x] + (-i*s)*in[x ^ 0x8000]
// ---------------------------------------------------------------------------
__global__ void k_q15perm(int layer, int sel) {
    const float2* in  = sel ? g_stB : g_stA;
    float2*       out = sel ? g_stA : g_stB;
    const int g = blockIdx.x * blockDim.x + threadIdx.x;
    const int s = g >> NQ;
    const unsigned y  = (unsigned)g & (NSTATE - 1);
    const unsigned t2 = y ^ (y << 1);
    const unsigned b15 = (y >> 15) & 1u;
    const unsigned x = (t2 & 0xFFFCu)
                     | (((((t2 >> 1) & 1u) ^ b15)) << 1)
                     | ((y & 1u) ^ b15);
    const float c  = g_cw[layer * NQ + 15];
    const float sn = g_sw[layer * NQ + 15];
    const float2 A  = in[(size_t)s * NSTATE + x];
    const float2 Bv = in[(size_t)s * NSTATE + (x ^ 0x8000u)];
    out[(size_t)s * NSTATE + y] =
        make_float2(c * A.x + sn * Bv.y, c * A.y - sn * Bv.x);
}

// ---------------------------------------------------------------------------
// 4) <Z_q> expvals: per-thread signed partial sums + LDS float atomics.
//    x = (j<<10)|tid : qubits 0..9 sign from tid, qubits 10..15 sign from j.
// ---------------------------------------------------------------------------
__global__ void k_expval() {
    const int s = blockIdx.x;
    const int tid = threadIdx.x;
    __shared__ float zl[NQ];
    if (tid < NQ) zl[tid] = 0.0f;
    __syncthreads();

    float T = 0.0f;
    float a6[6] = {0, 0, 0, 0, 0, 0};
#pragma unroll
    for (int j = 0; j < 64; ++j) {
        const float2 v = g_stA[(size_t)s * NSTATE + (j << 10) + tid];
        const float p = v.x * v.x + v.y * v.y;
        T += p;
#pragma unroll
        for (int b = 0; b < 6; ++b)
            a6[b] += ((j >> b) & 1) ? -p : p;
    }
#pragma unroll
    for (int q = 0; q < 10; ++q)
        atomicAdd(&zl[q], ((tid >> q) & 1) ? -T : T);
#pragma unroll
    for (int b = 0; b < 6; ++b)
        atomicAdd(&zl[10 + b], a6[b]);
    __syncthreads();
    if (tid < NQ) g_zq[s * NQ + tid] = zl[tid];
}

// ---------------------------------------------------------------------------
// 5) tanh(zq @ w_lin^T + b) via V_WMMA_F32_16X16X4_F32.
//    One wave per 16x16 output tile; K=16 in 4 chained WMMA steps.
//    A layout (ISA 16x4 f32): lanes 0-15 -> M=lane, K = k0+{0,1};
//                             lanes 16-31 -> M=lane-16, K = k0+{2,3}.
//    C layout: VGPR r -> M = r (+8 for lanes 16-31), N = lane%16.
// ---------------------------------------------------------------------------
__global__ void k_linear_wmma(const float* __restrict__ wlin,
                              const float* __restrict__ blin,
                              float* __restrict__ out) {
    const int m0   = blockIdx.x * 16;     // batch tile (64 rows -> 4 tiles)
    const int n0   = blockIdx.y * 16;     // feature tile (784 -> 49 tiles)
    const int lane = threadIdx.x;         // wave32
    const int half = lane >> 4;
    const int l16  = lane & 15;

    v8f acc = {};
#pragma unroll
    for (int k0 = 0; k0 < NQ; k0 += 4) {
        const int ka = k0 + 2 * half;
        v2f a, b;
        a.x = g_zq[(m0 + l16) * NQ + ka + 0];
        a.y = g_zq[(m0 + l16) * NQ + ka + 1];
        // B[k][n] = wlin[n][k]  (w_lin is row-major 784x16)
        b.x = wlin[(n0 + l16) * NQ + ka + 0];
        b.y = wlin[(n0 + l16) * NQ + ka + 1];
        acc = __builtin_amdgcn_wmma_f32_16x16x4_f32(
            /*neg_a=*/false, a, /*neg_b=*/false, b,
            /*c_mod=*/(short)0, acc, /*reuse_a=*/false, /*reuse_b=*/false);
    }

    const int n = n0 + l16;
    const float bias = blin[n];
    const int mbase = m0 + (half ? 8 : 0);
#pragma unroll
    for (int r = 0; r < 8; ++r) {
        out[(size_t)(mbase + r) * IMGPIX + n] = tanhf(acc[r] + bias);
    }
}

// ---------------------------------------------------------------------------
extern "C" void kernel_launch(void* const* d_in, const int* in_sizes, int n_in,
                              void* d_out, int out_size, void* d_ws, size_t ws_size,
                              hipStream_t stream) {
    (void)in_sizes; (void)n_in; (void)d_ws; (void)ws_size; (void)out_size;
    const float* z    = (const float*)d_in[0];   // (64,16)
    const float* qw   = (const float*)d_in[1];   // (4,16)
    const float* wlin = (const float*)d_in[2];   // (784,16)
    const float* blin = (const float*)d_in[3];   // (784,)
    float* out = (float*)d_out;                  // 64*784 fp32

    // Raise dynamic-LDS cap for the 256 KB butterfly kernel (capture-safe,
    // deterministic, called every launch).
    hipFuncSetAttribute(reinterpret_cast<const void*>(k_layer),
                        hipFuncAttributeMaxDynamicSharedMemorySize,
                        HALFST * (int)sizeof(float2));

    k_tables<<<1, 1024, 0, stream>>>(z, qw);
    k_init<<<NB, 1024, 0, stream>>>();
    for (int l = 0; l < QDEPTH; ++l) {
        const int sel = l & 1;   // state lives in A for even l, B for odd l
        k_layer<<<NB * 2, 1024, HALFST * sizeof(float2), stream>>>(l, sel);
        k_q15perm<<<(NB * NSTATE) / 256, 256, 0, stream>>>(l, sel);
    }
    // after 4 layers (even count) the state is back in g_stA
    k_expval<<<NB, 1024, 0, stream>>>();
    k_linear_wmma<<<dim3(4, 49), 32, 0, stream>>>(wlin, blin, out);
}